// AttentionalFactorizationMachineLayer_88158498718351
// MI455X (gfx1250) — compile-verified
//
#include <hip/hip_runtime.h>

typedef float v2f __attribute__((ext_vector_type(2)));
typedef float v8f __attribute__((ext_vector_type(8)));

#define NFIELDS 64
#define EDIM    64
#define ADIM    32
#define NPAIRS  2016      // 64*63/2
#define NTILES  126       // NPAIRS / 16
#define THREADS 256
#define NWAVES  8

__global__ __launch_bounds__(THREADS)
void afm_kernel(const float* __restrict__ emb,   // (B,64,64)
                const float* __restrict__ W1,    // (64,32)
                const float* __restrict__ b1,    // (32)
                const float* __restrict__ W2,    // (32,1)
                const float* __restrict__ b2,    // (1)
                float* __restrict__ out,         // (B,64)
                float* __restrict__ attn)        // (B,2016)
{
    __shared__ float          s_emb[NFIELDS * EDIM];   // 16 KB
    __shared__ float          s_w1[EDIM * ADIM];       // 8 KB
    __shared__ float          s_b1[ADIM];
    __shared__ float          s_w2[ADIM];
    __shared__ unsigned short s_pair[NPAIRS];          // ~4 KB
    __shared__ float          s_logit[NPAIRS];         // ~8 KB
    __shared__ float          s_h[NWAVES * 16 * ADIM]; // 16 KB
    __shared__ float          s_red[THREADS];          // 1 KB

    const int tid  = threadIdx.x;
    const int b    = blockIdx.x;
    const int wave = tid >> 5;
    const int lane = tid & 31;
    const int m    = lane & 15;    // row / column slot within 16
    const int half = lane >> 4;    // 0 or 1

    // ---- stage inputs into LDS ----
    const float* embb = emb + (size_t)b * (NFIELDS * EDIM);
    for (int i = tid; i < NFIELDS * EDIM; i += THREADS) s_emb[i] = embb[i];
    for (int i = tid; i < EDIM * ADIM;   i += THREADS) s_w1[i]  = W1[i];
    if (tid < ADIM) { s_b1[tid] = b1[tid]; s_w2[tid] = W2[tid]; }

    // ---- pair (i<j) enumeration table, torch triu order ----
    for (int p = tid; p < NPAIRS; p += THREADS) {
        int i = 0, rem = p;
        while (rem >= (NFIELDS - 1 - i)) { rem -= (NFIELDS - 1 - i); ++i; }
        int j = i + 1 + rem;
        s_pair[p] = (unsigned short)((i << 8) | j);
    }
    __syncthreads();

    // ---- preload W1 B-fragments (reused by every tile) ----
    // k-step k covers K = 4k..4k+3; this lane holds rows e0 = 4k+2*half, e0+1
    v2f bf[2][16];
    #pragma unroll
    for (int k = 0; k < 16; ++k) {
        const int e0 = 4 * k + 2 * half;
        #pragma unroll
        for (int n = 0; n < 2; ++n) {
            bf[n][k].x = s_w1[(e0)     * ADIM + 16 * n + m];
            bf[n][k].y = s_w1[(e0 + 1) * ADIM + 16 * n + m];
        }
    }
    const float bias2 = b2[0];
    float* hsc = s_h + wave * (16 * ADIM);

    // ---- per-tile: h = relu(inner@W1 + b1); logit = h@W2 + b2 ----
    for (int t = wave; t < NTILES; t += NWAVES) {
        const int pbase = t * 16;
        const int pr = s_pair[pbase + m];
        const int fi = pr >> 8, fj = pr & 255;
        const float* ei = s_emb + fi * EDIM;
        const float* ej = s_emb + fj * EDIM;

        v8f c0, c1;
        const float bi0 = s_b1[m], bi1 = s_b1[16 + m];
        #pragma unroll
        for (int r = 0; r < 8; ++r) { c0[r] = bi0; c1[r] = bi1; }

        #pragma unroll
        for (int k = 0; k < 16; ++k) {
            const int e0 = 4 * k + 2 * half;
            v2f a;
            a.x = ei[e0]     * ej[e0];
            a.y = ei[e0 + 1] * ej[e0 + 1];
            c0 = __builtin_amdgcn_wmma_f32_16x16x4_f32(false, a, false, bf[0][k],
                                                       (short)0, c0, false, false);
            c1 = __builtin_amdgcn_wmma_f32_16x16x4_f32(false, a, false, bf[1][k],
                                                       (short)0, c1, false, false);
        }

        // relu -> per-wave LDS scratch. Row M = r + 8*half, col = m / 16+m.
        #pragma unroll
        for (int r = 0; r < 8; ++r) {
            const int M = r + 8 * half;
            hsc[M * ADIM + m]      = fmaxf(c0[r], 0.0f);
            hsc[M * ADIM + 16 + m] = fmaxf(c1[r], 0.0f);
        }
        // lanes 0..15: 32-wide dot with W2 for this tile's 16 pairs
        if (half == 0) {
            float acc = bias2;
            #pragma unroll
            for (int aa = 0; aa < ADIM; ++aa) acc += hsc[m * ADIM + aa] * s_w2[aa];
            s_logit[pbase + m] = acc;
        }
    }
    __syncthreads();

    // ---- block-wide softmax over 2016 logits ----
    float lmax = -3.4e38f;
    for (int p = tid; p < NPAIRS; p += THREADS) lmax = fmaxf(lmax, s_logit[p]);
    s_red[tid] = lmax;
    __syncthreads();
    for (int s = THREADS / 2; s > 0; s >>= 1) {
        if (tid < s) s_red[tid] = fmaxf(s_red[tid], s_red[tid + s]);
        __syncthreads();
    }
    const float gmax = s_red[0];
    __syncthreads();

    float lsum = 0.0f;
    for (int p = tid; p < NPAIRS; p += THREADS) {
        const float e = expf(s_logit[p] - gmax);
        s_logit[p] = e;
        lsum += e;
    }
    s_red[tid] = lsum;
    __syncthreads();
    for (int s = THREADS / 2; s > 0; s >>= 1) {
        if (tid < s) s_red[tid] += s_red[tid + s];
        __syncthreads();
    }
    const float inv = 1.0f / s_red[0];
    __syncthreads();

    // attn_scores output
    float* attnb = attn + (size_t)b * NPAIRS;
    for (int p = tid; p < NPAIRS; p += THREADS) attnb[p] = s_logit[p] * inv;

    // ---- out[e] = sum_p attn_p * emb[i_p][e]*emb[j_p][e] ----
    const int e = tid & 63;
    const int g = tid >> 6;              // 4 groups of 64 lanes
    float acc = 0.0f;
    for (int p = g; p < NPAIRS; p += 4) {
        const int pr = s_pair[p];
        const int fi = pr >> 8, fj = pr & 255;
        acc += s_logit[p] * s_emb[fi * EDIM + e] * s_emb[fj * EDIM + e];
    }
    s_red[g * 64 + e] = acc;
    __syncthreads();
    if (tid < 64) {
        out[(size_t)b * EDIM + tid] =
            (s_red[tid] + s_red[64 + tid] + s_red[128 + tid] + s_red[192 + tid]) * inv;
    }
}

extern "C" void kernel_launch(void* const* d_in, const int* in_sizes, int n_in,
                              void* d_out, int out_size, void* d_ws, size_t ws_size,
                              hipStream_t stream) {
    (void)in_sizes; (void)n_in; (void)d_ws; (void)ws_size; (void)out_size;
    const float* emb = (const float*)d_in[0];   // (1024,64,64)
    const float* W1  = (const float*)d_in[1];   // (64,32)
    const float* b1  = (const float*)d_in[2];   // (32)
    const float* W2  = (const float*)d_in[3];   // (32,1)
    const float* b2  = (const float*)d_in[4];   // (1)
    // d_in[5] = num_fields (python scalar) — fixed 64 here

    float* out  = (float*)d_out;                // outputs (1024,64) first
    float* attn = out + (size_t)1024 * 64;      // then attn_scores (1024,2016,1)

    afm_kernel<<<1024, THREADS, 0, stream>>>(emb, W1, b1, W2, b2, out, attn);
}